// LocalizationLoss_91070486544852
// MI455X (gfx1250) — compile-verified
//
#include <hip/hip_runtime.h>
#include <cstdint>

// ---- problem constants (match reference) ----
#define ROWLEN     1004        // 4 box preds + 1000 logits, row pitch = 4016 B (16B aligned)
#define NCLS       1000
#define TILE_ROWS  8           // rows per TDM tile == waves per block
#define WAVES      8
#define THREADS    256
#define ROWS_PER_WG 128
#define NTILES     (ROWS_PER_WG / TILE_ROWS)   // 16

typedef unsigned int uint32x4 __attribute__((ext_vector_type(4)));
typedef int          int32x8  __attribute__((ext_vector_type(8)));
typedef int          int32x4  __attribute__((ext_vector_type(4)));

// Issue a TDM load of a 2D tile (nrows x ROWLEN fp32, row stride ROWLEN) from
// global memory into LDS at byte offset lds_off. D# bitfields per CDNA5 ISA §8.
// This toolchain exposes the 6-arg builtin: (g0, g1, g2, g3, extra, cpol).
__device__ __forceinline__ void tdm_load_rows(const float* gsrc, unsigned lds_off, int nrows) {
  uint64_t ga = (uint64_t)(uintptr_t)gsrc;
  uint32x4 g0;
  g0[0] = 1u;                                             // count=1 (valid), user mode
  g0[1] = lds_off;                                        // lds_addr (bytes)
  g0[2] = (unsigned)(ga & 0xffffffffu);                   // global_addr[31:0]
  g0[3] = (unsigned)((ga >> 32) & 0x01ffffffu)            // global_addr[56:32]
        | (2u << 30);                                     // type = 2 ("image")
  int32x8 g1;
  g1[0] = (2 << 16);                                      // data_size = 2 (4 bytes); mask=0
  g1[1] = (int)((unsigned)(ROWLEN & 0xffff) << 16);       // tensor_dim0[15:0] -> bits 63:48
  g1[2] = (int)((unsigned)(nrows  & 0xffff) << 16);       // tensor_dim0 hi=0 | tensor_dim1[15:0]
  g1[3] = (int)((unsigned)(ROWLEN & 0xffff) << 16);       // tensor_dim1 hi=0 | tile_dim0
  g1[4] = (nrows & 0xffff);                               // tile_dim1 | tile_dim2=0
  g1[5] = ROWLEN;                                         // tensor_dim0_stride[31:0]
  g1[6] = 0;                                              // stride0 hi | dim1_stride lo
  g1[7] = 0;                                              // dim1_stride hi
  int32x4 z4 = {0, 0, 0, 0};                              // groups 2/3: unused dims
  int32x8 z8 = {0, 0, 0, 0, 0, 0, 0, 0};                  // trailing descriptor words
  __builtin_amdgcn_tensor_load_to_lds(g0, g1, z4, z4, z8, /*cpol=*/0);
}

__global__ __launch_bounds__(THREADS)
void loc_loss_stage1(const float* __restrict__ outp,
                     const float* __restrict__ tgt,
                     float* __restrict__ partial) {
  __shared__ __align__(16) float tile[2][TILE_ROWS][ROWLEN];   // 2 x 32128 B = 64256 B
  __shared__ float red[WAVES];

  const int wave = threadIdx.x >> 5;
  const int lane = threadIdx.x & 31;
  const long rowbase = (long)blockIdx.x * ROWS_PER_WG;

  // prologue: prefetch tile 0 into buffer 0 (wave 0 owns the TDM + TENSORcnt)
  if (wave == 0) {
    tdm_load_rows(outp + rowbase * ROWLEN,
                  (unsigned)(uintptr_t)&tile[0][0][0], TILE_ROWS);
  }

  float acc = 0.0f;

  for (int t = 0; t < NTILES; ++t) {
    if (wave == 0) __builtin_amdgcn_s_wait_tensorcnt(0);   // tile t resident in LDS
    __syncthreads();                                       // publishes tile t; prev compute done
    if (wave == 0 && (t + 1) < NTILES) {                   // double-buffered prefetch of t+1
      tdm_load_rows(outp + (rowbase + (long)(t + 1) * TILE_ROWS) * ROWLEN,
                    (unsigned)(uintptr_t)&tile[(t + 1) & 1][0][0], TILE_ROWS);
    }

    const long r = rowbase + (long)t * TILE_ROWS + wave;   // one wave per row
    const float* lrow = &tile[t & 1][wave][0];
    const int cls = (int)tgt[r * 5 + 4];

    // each lane holds 8 float4 of logits: columns 4*lane + 128*k (+0..3)
    float4 v[8];
#pragma unroll
    for (int k = 0; k < 8; ++k) {
      if (k < 7 || lane < 26) {
        v[k] = *reinterpret_cast<const float4*>(lrow + 4 + lane * 4 + k * 128);
      } else {
        v[k] = make_float4(-3.0e38f, -3.0e38f, -3.0e38f, -3.0e38f);  // padding slots
      }
    }

    // pass 1: row max
    float m = -3.0e38f;
#pragma unroll
    for (int k = 0; k < 8; ++k)
      m = fmaxf(m, fmaxf(fmaxf(v[k].x, v[k].y), fmaxf(v[k].z, v[k].w)));
#pragma unroll
    for (int off = 16; off >= 1; off >>= 1)
      m = fmaxf(m, __shfl_xor(m, off, 32));

    // pass 2: sum of exp + gather target-class logit
    float s = 0.0f, pick = 0.0f;
#pragma unroll
    for (int k = 0; k < 8; ++k) {
      const int j = lane * 4 + k * 128;
      s += expf(v[k].x - m); s += expf(v[k].y - m);
      s += expf(v[k].z - m); s += expf(v[k].w - m);
      if (j + 0 == cls) pick += v[k].x;
      if (j + 1 == cls) pick += v[k].y;
      if (j + 2 == cls) pick += v[k].z;
      if (j + 3 == cls) pick += v[k].w;
    }
#pragma unroll
    for (int off = 16; off >= 1; off >>= 1) {
      s    += __shfl_xor(s, off, 32);
      pick += __shfl_xor(pick, off, 32);
    }
    const float class_loss = logf(s) + m - pick;

    // box regression loss (uniform within wave; broadcast LDS/global reads)
    const float4 p = *reinterpret_cast<const float4*>(lrow);
    const float xmin = tgt[r * 5 + 0], ymin = tgt[r * 5 + 1];
    const float xmax = tgt[r * 5 + 2], ymax = tgt[r * 5 + 3];
    const float dx = p.x - (xmin + xmax) * 0.5f;
    const float dy = p.y - (ymin + ymax) * 0.5f;
    const float dw = p.z - (xmax - xmin);
    const float dh = p.w - (ymax - ymin);
    const float loc = 0.25f * (dx * dx + dy * dy + dw * dw + dh * dh);

    if (lane == 0) acc += loc + class_loss;
  }

  if (lane == 0) red[wave] = acc;
  __syncthreads();
  if (threadIdx.x == 0) {
    float s = 0.0f;
#pragma unroll
    for (int w = 0; w < WAVES; ++w) s += red[w];
    partial[blockIdx.x] = s;
  }
}

__global__ __launch_bounds__(256)
void loc_loss_stage2(const float* __restrict__ partial, int n, float scale,
                     float* __restrict__ outp) {
  __shared__ float red[256];
  float s = 0.0f;
  for (int i = threadIdx.x; i < n; i += 256) s += partial[i];
  red[threadIdx.x] = s;
  __syncthreads();
  for (int off = 128; off >= 1; off >>= 1) {
    if ((int)threadIdx.x < off) red[threadIdx.x] += red[threadIdx.x + off];
    __syncthreads();
  }
  if (threadIdx.x == 0) outp[0] = red[0] * scale;
}

extern "C" void kernel_launch(void* const* d_in, const int* in_sizes, int n_in,
                              void* d_out, int out_size, void* d_ws, size_t ws_size,
                              hipStream_t stream) {
  const float* outp = (const float*)d_in[0];   // (B, 1004) f32
  const float* tgt  = (const float*)d_in[1];   // (B, 5)    f32
  const int rows = in_sizes[1] / 5;            // B = 262144
  const int grid = rows / ROWS_PER_WG;         // 2048 blocks
  float* partial = (float*)d_ws;               // grid * 4 bytes of scratch

  loc_loss_stage1<<<grid, THREADS, 0, stream>>>(outp, tgt, partial);
  loc_loss_stage2<<<1, 256, 0, stream>>>(partial, grid, 1.0f / (float)rows,
                                         (float*)d_out);
}